// _Volume4dImpl_65085934403719
// MI455X (gfx1250) — compile-verified
//
#include <hip/hip_runtime.h>

// 4-D quadrilinear interpolation on a 64^4 grid with 4 channels.
// data: [64][64][64][64][4] f32 (each cell = one float4, 16 B, contiguous)
// xs:   [N][4] f32 in [-1, 1]
// out:  [N][4] f32
//
// Memory-gather bound (AI ~0.3 FLOP/B): no matrix structure, so WMMA does not
// apply. Optimization target is the VMEM path:
//   - 16 outstanding global_load_b128 gathers per lane (MLP)
//   - 32-bit element indices -> GVS addressing with scale_offset (no 64-bit
//     VALU address math)
//   - NT hints on streaming xs/out so the 192 MB L2 stays dedicated to the
//     256 MB table
//   - global_prefetch_b8 on the xs stream (gfx1250 prefetch path)

typedef float v4f __attribute__((ext_vector_type(4)));

namespace {
constexpr int kS   = 64;         // grid size per axis
constexpr int kSm1 = kS - 1;
constexpr int kN   = 4194304;    // number of query points (multiple of kBlk)
constexpr int kBlk = 256;        // 8 wave32 per block
}

__global__ __launch_bounds__(kBlk) void volume4d_lerp_kernel(
    const v4f* __restrict__ grid,   // 64^4 cells, cell = float4 channels
    const v4f* __restrict__ xs,     // query points
    v4f* __restrict__ out)          // outputs
{
  const int n = blockIdx.x * kBlk + threadIdx.x;   // grid is exact, no guard

  // Streaming input: non-temporal so it does not evict table lines from L2.
  const v4f p = __builtin_nontemporal_load(&xs[n]);

  // Prefetch the xs stream ahead (emits global_prefetch_b8 on gfx1250).
  if (n + 16384 < kN) __builtin_prefetch((const void*)&xs[n + 16384], 0, 0);

  // Map [-1,1] -> [0,64) grid coordinates (matches reference exactly).
  float c[4];
  c[0] = (0.5f * p.x + 0.5f) * (float)kS;
  c[1] = (0.5f * p.y + 0.5f) * (float)kS;
  c[2] = (0.5f * p.z + 0.5f) * (float)kS;
  c[3] = (0.5f * p.w + 0.5f) * (float)kS;

  int   lo[4], hi[4];
  float w0[4], w1[4];
#pragma unroll
  for (int a = 0; a < 4; ++a) {
    const int   idx = (int)c[a];             // trunc toward zero (astype int32)
    const float f   = c[a] - (float)idx;     // lerp weight from UNCLAMPED idx
    int l = idx;
    l = l < 0    ? 0    : l;
    l = l > kSm1 ? kSm1 : l;
    int h = l + 1;
    h = h > kSm1 ? kSm1 : h;
    lo[a] = l;  hi[a] = h;
    w1[a] = f;  w0[a] = 1.0f - f;
  }

  // Factorized 32-bit element indices: idx = xy + zt, each corner = 1 v_add.
  // Element index max = 64^4-1 = 2^24-1; *16 B < 2^31 -> GVS scale_offset OK.
  const int xs0 = lo[0] << 18, xs1 = hi[0] << 18;
  const int ys0 = lo[1] << 12, ys1 = hi[1] << 12;
  const int zs0 = lo[2] << 6,  zs1 = hi[2] << 6;
  const int t0  = lo[3],       t1  = hi[3];

  const int xy[4] = {xs0 + ys0, xs0 + ys1, xs1 + ys0, xs1 + ys1};
  const int zt[4] = {zs0 + t0,  zs0 + t1,  zs1 + t0,  zs1 + t1};

  const float wxy[4] = {w0[0] * w0[1], w0[0] * w1[1],
                        w1[0] * w0[1], w1[0] * w1[1]};
  const float wzt[4] = {w0[2] * w0[3], w0[2] * w1[3],
                        w1[2] * w0[3], w1[2] * w1[3]};

  // Issue all 16 b128 gathers up front -> 16 outstanding loads per lane.
  // Corner order k = (bx,by,bz,bt) matches the reference accumulation order
  // so floating-point summation order is identical.
  v4f   v[16];
  float w[16];
#pragma unroll
  for (int i = 0; i < 4; ++i) {        // i = bx*2+by
#pragma unroll
    for (int j = 0; j < 4; ++j) {      // j = bz*2+bt
      const int k = i * 4 + j;
      v[k] = grid[xy[i] + zt[j]];      // global_load_b128, saddr+voffset form
      w[k] = wxy[i] * wzt[j];
    }
  }

  v4f acc = (v4f)(0.0f);
#pragma unroll
  for (int k = 0; k < 16; ++k) {
    acc += v[k] * w[k];                // contracts to v_fma per channel
  }

  // Streaming output: non-temporal store to protect L2 table residency.
  __builtin_nontemporal_store(acc, &out[n]);
}

extern "C" void kernel_launch(void* const* d_in, const int* in_sizes, int n_in,
                              void* d_out, int out_size, void* d_ws, size_t ws_size,
                              hipStream_t stream) {
  (void)in_sizes; (void)n_in; (void)out_size; (void)d_ws; (void)ws_size;
  const v4f* grid = (const v4f*)d_in[0];   // data: 64^4 * 4 f32
  const v4f* xs   = (const v4f*)d_in[1];   // xs:   N * 4 f32
  v4f*       out  = (v4f*)d_out;           // out:  N * 4 f32

  const int blocks = kN / kBlk;            // 16384
  volume4d_lerp_kernel<<<blocks, kBlk, 0, stream>>>(grid, xs, out);
}